// InstantSpeech_36799279792712
// MI455X (gfx1250) — compile-verified
//
#include <hip/hip_runtime.h>

#define MAX_FRAMES 2048
#define BB 16
#define LL 256
#define DD 512

typedef __attribute__((ext_vector_type(2))) float v2f;
typedef __attribute__((ext_vector_type(8))) float v8f;
typedef __attribute__((ext_vector_type(4))) unsigned int v4u;
typedef __attribute__((ext_vector_type(4))) int v4i;
typedef __attribute__((ext_vector_type(8))) int v8i;

#if defined(__has_builtin)
#if __has_builtin(__builtin_amdgcn_wmma_f32_16x16x4_f32)
#define HAVE_WMMA_F32X4 1
#endif
#if __has_builtin(__builtin_amdgcn_tensor_load_to_lds) && \
    __has_builtin(__builtin_amdgcn_s_wait_tensorcnt)
#define HAVE_TDM 1
#endif
#endif

// ---------------------------------------------------------------------------
// Kernel 1: per-batch duration inclusive scan + nonzero-mean pitch per symbol
// ---------------------------------------------------------------------------
__global__ __launch_bounds__(LL) void prep_kernel(
    const float* __restrict__ pitch, const int* __restrict__ durs,
    int* __restrict__ g_ends, float* __restrict__ g_pa) {
  const int b = blockIdx.x;
  const int l = threadIdx.x;
  __shared__ int s_scan[LL];
  const int dur = durs[b * LL + l];
  s_scan[l] = dur;
  __syncthreads();
  for (int off = 1; off < LL; off <<= 1) {
    int v = (l >= off) ? s_scan[l - off] : 0;
    __syncthreads();
    s_scan[l] += v;
    __syncthreads();
  }
  const int end = s_scan[l];
  const int start = end - dur;
  const float* p = pitch + (size_t)b * MAX_FRAMES;
  float sum = 0.f, cnt = 0.f;
  for (int f = start; f < end; ++f) {
    float v = p[f];
    sum += v;
    cnt += (v != 0.f) ? 1.f : 0.f;
  }
  g_ends[b * LL + l] = end;
  g_pa[b * LL + l] = (cnt == 0.f) ? 0.f : sum / cnt;
}

// ---------------------------------------------------------------------------
// Kernel 2: TDM row-gather into LDS + conv1d(k=3) via WMMA f32 16x16x4
// grid = B * (MAX_FRAMES/16) blocks, 256 threads = 8 waves.
// ---------------------------------------------------------------------------
__global__ __launch_bounds__(256) void expand_kernel(
    const float* __restrict__ enc, const float* __restrict__ w,
    const float* __restrict__ bias, const int* __restrict__ g_ends,
    const float* __restrict__ g_pa, float* __restrict__ out) {
  const int tid = threadIdx.x;
  const int b = blockIdx.x >> 7;            // / (MAX_FRAMES/16)
  const int t0 = (blockIdx.x & 127) << 4;   // frame tile origin

  __shared__ int s_total;
  if (tid == 0) s_total = g_ends[b * LL + LL - 1];
  __syncthreads();
  const int total = s_total;

  float* outB = out + ((size_t)b * MAX_FRAMES + t0) * DD;

  // Uniform fast path: whole tile past the last frame -> pure zero fill.
  if (t0 >= total) {
    float4 z = make_float4(0.f, 0.f, 0.f, 0.f);
    float4* o4 = (float4*)outB;
#pragma unroll
    for (int i = 0; i < 8; ++i) o4[tid + i * 256] = z;
    return;
  }

  __shared__ int   s_ends[LL];
  __shared__ float s_pa[LL + 2];                    // SAME-padding halo
  __shared__ __align__(16) float s_wp[DD * 4];      // {w0,w1,w2,0} per channel
  __shared__ __align__(16) float s_pap[LL * 4];     // {pa[l-1],pa[l],pa[l+1],0}
  __shared__ float s_bias[DD];
  __shared__ int   s_lsym[16];
  __shared__ __align__(16) float s_enc[16 * DD];    // gathered enc rows (32 KB)

  s_ends[tid] = g_ends[b * LL + tid];
  s_pa[tid + 1] = g_pa[b * LL + tid];
  if (tid == 0) { s_pa[0] = 0.f; s_pa[LL + 1] = 0.f; }
  s_bias[tid] = bias[tid];
  s_bias[tid + 256] = bias[tid + 256];
#pragma unroll
  for (int q = 0; q < 2; ++q) {                     // K=3 padded to 4 (zero)
    const int d = tid + q * 256;
    s_wp[d * 4 + 0] = w[d * 3 + 0];
    s_wp[d * 4 + 1] = w[d * 3 + 1];
    s_wp[d * 4 + 2] = w[d * 3 + 2];
    s_wp[d * 4 + 3] = 0.f;
  }
  __syncthreads();

  // Pair-friendly pitch windows (branchless v2f loads later).
  s_pap[tid * 4 + 0] = s_pa[tid];
  s_pap[tid * 4 + 1] = s_pa[tid + 1];
  s_pap[tid * 4 + 2] = s_pa[tid + 2];
  s_pap[tid * 4 + 3] = 0.f;

  // Frame -> symbol: first l with ends[l] > t. Invalid frames clamp to LL-1
  // (keeps the TDM gather index list non-decreasing; content masked later).
  if (tid < 16) {
    const int t = t0 + tid;
    int ls = LL - 1;
    if (t < total) {
      int lo = 0, hi = LL - 1;
      while (lo < hi) {
        int mid = (lo + hi) >> 1;
        if (s_ends[mid] > t) hi = mid; else lo = mid + 1;
      }
      ls = lo;
      __builtin_prefetch(enc + ((size_t)b * LL + ls) * DD, 0, 0);
    }
    s_lsym[tid] = ls;
  }
  __syncthreads();

#ifdef HAVE_TDM
  // One TDM gather: 16 rows of enc[b] (row index = s_lsym[M]) -> s_enc.
  if ((tid >> 5) == 0) {
    const unsigned long long gaddr =
        (unsigned long long)(uintptr_t)(enc + (size_t)b * LL * DD);
    const unsigned lds_off = (unsigned)(uintptr_t)&s_enc[0];
    v4u g0;
    g0[0] = 1u | (1u << 31);                       // count=1, gather_mode=1, 16-bit idx
    g0[1] = lds_off;                               // LDS dest (bytes)
    g0[2] = (unsigned)(gaddr & 0xFFFFFFFFu);       // global_addr[31:0]
    g0[3] = (unsigned)((gaddr >> 32) & 0x01FFFFFFu) | (2u << 30); // addr[56:32], type=2
    v8i g1;
    g1[0] = (int)(2u << 16);                       // data_size = 4B
    g1[1] = (int)((unsigned)DD << 16);             // tensor_dim0[15:0]
    g1[2] = (int)((unsigned)LL << 16);             // tensor_dim1[15:0]
    g1[3] = (int)((unsigned)DD << 16);             // tile_dim0 = 512
    g1[4] = 16;                                    // tile_dim1 = #indices
    g1[5] = DD;                                    // tensor_dim0_stride[31:0]
    g1[6] = 0;
    g1[7] = 0;
    v4i g2, g3;
#pragma unroll
    for (int j = 0; j < 4; ++j) {
      g2[j] = (s_lsym[2 * j] & 0xFFFF) | (s_lsym[2 * j + 1] << 16);
      g3[j] = (s_lsym[8 + 2 * j] & 0xFFFF) | (s_lsym[8 + 2 * j + 1] << 16);
    }
    v8i g4 = {0, 0, 0, 0, 0, 0, 0, 0};             // extra group (6-arg form): zero
    __builtin_amdgcn_tensor_load_to_lds(g0, g1, g2, g3, g4, 0);
    __builtin_amdgcn_s_wait_tensorcnt((short)0);
  }
  __syncthreads();
#else
  // Fallback: cooperative coalesced float4 staging of the 16 gathered rows.
  {
    float4* s4 = (float4*)s_enc;
    const float4* e4 = (const float4*)(enc + (size_t)b * LL * DD);
#pragma unroll
    for (int i = tid; i < 16 * (DD / 4); i += 256) {
      int M = i >> 7;             // row (DD/4 = 128 float4 per row)
      int c4 = i & 127;
      s4[i] = e4[(size_t)s_lsym[M] * (DD / 4) + c4];
    }
  }
  __syncthreads();
#endif

  const int wave = tid >> 5;
  const int lane = tid & 31;
  const int n    = lane & 15;
  const int hi16 = lane >> 4;
  const int k0   = hi16 << 1;     // K base this lane group holds: 0 or 2

  // Branchless validity masks (pure ALU, no LDS, no exec divergence).
  float vm[8];
#pragma unroll
  for (int r = 0; r < 8; ++r)
    vm[r] = (t0 + r + (hi16 << 3) < total) ? 1.f : 0.f;

  // A fragment: one aligned b64 LDS load, scaled by validity; K=3 slot is 0.
  const int ls = s_lsym[n];
  const float av = (t0 + n < total) ? 1.f : 0.f;
  v2f afrag = *(const v2f*)&s_pap[ls * 4 + k0];
  afrag.x *= av;
  afrag.y *= av;

#pragma unroll
  for (int i = 0; i < 4; ++i) {
    const int d0 = (wave + i * 8) << 4;
    const int d  = d0 + n;
#ifdef HAVE_WMMA_F32X4
    // B fragment: one aligned b64 LDS load (k0=0 -> {w0,w1}; k0=2 -> {w2,0}).
    v2f bfrag = *(const v2f*)&s_wp[d * 4 + k0];
    v8f cfrag;
#pragma unroll
    for (int r = 0; r < 8; ++r) {
      const int M = r + (hi16 << 3);
      cfrag[r] = vm[r] * (s_enc[M * DD + d] + s_bias[d]);
    }
    v8f dfrag = __builtin_amdgcn_wmma_f32_16x16x4_f32(
        false, afrag, false, bfrag, (short)0, cfrag, false, false);
#pragma unroll
    for (int r = 0; r < 8; ++r) {
      const int M = r + (hi16 << 3);
      outB[(size_t)M * DD + d] = dfrag[r];
    }
#else
#pragma unroll
    for (int r = 0; r < 8; ++r) {
      const int M = r + (hi16 << 3);
      const int lsr = s_lsym[M];
      float v = vm[r] * (s_enc[M * DD + d] + s_bias[d]
                         + s_pap[lsr * 4 + 0] * s_wp[d * 4 + 0]
                         + s_pap[lsr * 4 + 1] * s_wp[d * 4 + 1]
                         + s_pap[lsr * 4 + 2] * s_wp[d * 4 + 2]);
      outB[(size_t)M * DD + d] = v;
    }
#endif
  }
}

extern "C" void kernel_launch(void* const* d_in, const int* in_sizes, int n_in,
                              void* d_out, int out_size, void* d_ws, size_t ws_size,
                              hipStream_t stream) {
  const float* enc   = (const float*)d_in[0];   // [B,L,D]
  const float* pitch = (const float*)d_in[1];   // [B,1,2048]
  const float* w     = (const float*)d_in[2];   // [D,1,3]
  const float* bias  = (const float*)d_in[3];   // [D]
  const int*   durs  = (const int*)d_in[4];     // [B,L]
  float* out = (float*)d_out;                   // [B,2048,D]

  int*   g_ends = (int*)d_ws;
  float* g_pa   = (float*)((char*)d_ws + BB * LL * sizeof(int));

  prep_kernel<<<BB, LL, 0, stream>>>(pitch, durs, g_ends, g_pa);
  expand_kernel<<<BB * (MAX_FRAMES / 16), 256, 0, stream>>>(
      enc, w, bias, g_ends, g_pa, out);
}